// TemporalAttention_48954037240016
// MI455X (gfx1250) — compile-verified
//
#include <hip/hip_runtime.h>
#include <math.h>

// ---------------- problem constants ----------------
#define BB   16
#define TT   48
#define NN   500
#define DD   64
#define HH   8
#define KH   8
#define MTOT (BB*TT*NN)          // 384000 rows
#define NEGF (-32767.0f)

typedef float v2f __attribute__((ext_vector_type(2)));
typedef float v8f __attribute__((ext_vector_type(8)));

// fp32 WMMA: D(16x16,f32) = A(16x4,f32) * B(4x16,f32) + C
// A frag (v2f): lanes 0-15 -> {A[M=l][k0], A[M=l][k0+1]}, lanes 16-31 -> {A[M=l-16][k0+2], A[M=l-16][k0+3]}
// B frag (v2f): lanes 0-15 -> {B[k0][N=l], B[k0+1][N=l]},  lanes 16-31 -> {B[k0+2][N=l-16], B[k0+3][N=l-16]}
// C/D (v8f):    VGPR r, lanes 0-15 -> row r, lanes 16-31 -> row r+8; col = lane&15
static __device__ __forceinline__ v8f wmma_f32(v2f a, v2f b, v8f c) {
    return __builtin_amdgcn_wmma_f32_16x16x4_f32(false, a, false, b, (short)0, c, false, false);
}

// stats layout in ws (floats), base = statp:
//  [0,192)   sum  q|k|v      [192,384) sumsq q|k|v
//  [384,576) a (scale) q|k|v [576,768) c (shift) q|k|v
//  [768,832) sum o  [832,896) sumsq o  [896,960) a o  [960,1024) c o

__global__ void zero_stats_kernel(float* __restrict__ stat) {
    for (int i = threadIdx.x; i < 1024; i += 256) stat[i] = 0.0f;
}

// ---------------- stage 1: y_{q,k,v} = concat(X,STE) @ W^T, per-channel stats ----------------
__global__ __launch_bounds__(128)
void fc_qkv_kernel(const float* __restrict__ X, const float* __restrict__ STE,
                   const float* __restrict__ Wq, const float* __restrict__ Wk,
                   const float* __restrict__ Wv,
                   float* __restrict__ yq, float* __restrict__ yk, float* __restrict__ yv,
                   float* __restrict__ stat)
{
    __shared__ float red[384];
    const int tid  = threadIdx.x;
    const int wave = tid >> 5;
    const int lane = tid & 31;
    for (int i = tid; i < 384; i += 128) red[i] = 0.0f;
    __syncthreads();

    const int m0    = (blockIdx.x * 4 + wave) * 16;
    const int row   = lane & 15;          // M row in A frag / N col in B,C frags
    const int khalf = (lane >> 4) * 2;    // K sub-offset per lane half
    const int rowhi = (lane >> 4) * 8;    // C frag row offset per lane half

    v8f acc[12];
    #pragma unroll
    for (int i = 0; i < 12; i++)
        #pragma unroll
        for (int e = 0; e < 8; e++) acc[i][e] = 0.0f;

    const size_t mrow = (size_t)(m0 + row);
    const float* Wm[3] = {Wq, Wk, Wv};

    for (int kk = 0; kk < 128; kk += 4) {
        const int c = kk + khalf;               // even; c and c+1 on same side of 64
        const float* src = (c < 64) ? (X + mrow * 64 + c) : (STE + mrow * 64 + (c - 64));
        v2f a; a.x = src[0]; a.y = src[1];
        #pragma unroll
        for (int mat = 0; mat < 3; mat++) {
            const float* W = Wm[mat];
            #pragma unroll
            for (int nt = 0; nt < 4; nt++) {
                const int o = nt * 16 + row;
                const float* wp = W + o * 128 + kk + khalf;
                v2f b; b.x = wp[0]; b.y = wp[1];
                acc[mat * 4 + nt] = wmma_f32(a, b, acc[mat * 4 + nt]);
            }
        }
    }

    float* ybuf[3] = {yq, yk, yv};
    #pragma unroll
    for (int mat = 0; mat < 3; mat++) {
        #pragma unroll
        for (int nt = 0; nt < 4; nt++) {
            const int col = nt * 16 + row;
            float s = 0.0f, ss = 0.0f;
            #pragma unroll
            for (int r = 0; r < 8; r++) {
                const float v = acc[mat * 4 + nt][r];
                ybuf[mat][(size_t)(m0 + r + rowhi) * 64 + col] = v;
                s += v; ss += v * v;
            }
            atomicAdd(&red[mat * 64 + col], s);
            atomicAdd(&red[192 + mat * 64 + col], ss);
        }
    }
    __syncthreads();
    for (int i = tid; i < 384; i += 128) atomicAdd(&stat[i], red[i]);
}

// ---------------- BN finalize: a = gamma*rsqrt(var+eps), c = beta - a*mu ----------------
__global__ void bn_finalize_kernel(const float* __restrict__ sum, const float* __restrict__ sumsq,
                                   const float* __restrict__ gamma, const float* __restrict__ beta,
                                   float* __restrict__ a, float* __restrict__ c, float invM)
{
    const int i = threadIdx.x;
    if (i < 64) {
        const float mu  = sum[i] * invM;
        const float var = sumsq[i] * invM - mu * mu;
        const float ai  = gamma[i] * rsqrtf(var + 1e-5f);
        a[i] = ai;
        c[i] = beta[i] - ai * mu;
    }
}

// ---------------- stage 3: causal attention per (b,n); 2 waves, 4 heads each ----------------
__global__ __launch_bounds__(64)
void attn_kernel(const float* yq, const float* __restrict__ yk,
                 const float* __restrict__ yv, const float* __restrict__ stat,
                 float* yatt /* may alias yq: all reads land in LDS before writes */)
{
    __shared__ float q_s[HH * TT * KH];   // [h][t][k]  12 KB
    __shared__ float k_s[HH * TT * KH];
    __shared__ float v_s[HH * TT * KH];
    __shared__ float P[2 * TT * TT];      // per-wave 48x48 scores  18 KB

    const int tid  = threadIdx.x;
    const int wave = tid >> 5;
    const int lane = tid & 31;
    const int b    = blockIdx.x / NN;
    const int n    = blockIdx.x % NN;

    const float* aP = stat + 384;   // q:[0,64) k:[64,128) v:[128,192)
    const float* cP = stat + 576;

    for (int idx = tid; idx < TT * DD; idx += 64) {
        const int t  = idx >> 6;
        const int ch = idx & 63;
        const size_t m = ((size_t)(b * TT + t) * NN + n) * 64 + ch;
        const int l = (ch >> 3) * (TT * KH) + t * KH + (ch & 7);
        q_s[l] = fmaxf(0.0f, aP[ch]       * yq[m] + cP[ch]);
        k_s[l] = fmaxf(0.0f, aP[64 + ch]  * yk[m] + cP[64 + ch]);
        v_s[l] = fmaxf(0.0f, aP[128 + ch] * yv[m] + cP[128 + ch]);
    }
    __syncthreads();

    const int row = lane & 15;
    const int hi  = lane >> 4;
    float* Pw = P + wave * (TT * TT);
    const float scale = 0.35355339059327373f;  // 1/sqrt(8)

    for (int hh = 0; hh < 4; hh++) {
        const int h = wave * 4 + hh;
        const float* qh = q_s + h * (TT * KH);
        const float* kh = k_s + h * (TT * KH);
        const float* vh = v_s + h * (TT * KH);

        // scores = q @ k^T / sqrt(d), causal-masked, into Pw
        #pragma unroll
        for (int i = 0; i < 3; i++)
            #pragma unroll
            for (int j = 0; j < 3; j++) {
                v8f acc;
                #pragma unroll
                for (int e = 0; e < 8; e++) acc[e] = 0.0f;
                #pragma unroll
                for (int kk = 0; kk < 8; kk += 4) {
                    const int k0 = kk + hi * 2;
                    v2f a;  a.x  = qh[(16 * i + row) * KH + k0]; a.y = qh[(16 * i + row) * KH + k0 + 1];
                    v2f bb; bb.x = kh[(16 * j + row) * KH + k0]; bb.y = kh[(16 * j + row) * KH + k0 + 1];
                    acc = wmma_f32(a, bb, acc);
                }
                #pragma unroll
                for (int r = 0; r < 8; r++) {
                    const int t = 16 * i + r + hi * 8;
                    const int s = 16 * j + row;
                    float v = acc[r] * scale;
                    if (s > t) v = NEGF;
                    Pw[t * TT + s] = v;
                }
            }
        __syncthreads();

        // row softmax
        for (int t = lane; t < TT; t += 32) {
            float mx = -3.4e38f;
            for (int s = 0; s < TT; s++) mx = fmaxf(mx, Pw[t * TT + s]);
            float sum = 0.0f;
            for (int s = 0; s < TT; s++) {
                const float e = expf(Pw[t * TT + s] - mx);
                Pw[t * TT + s] = e; sum += e;
            }
            const float inv = 1.0f / sum;
            for (int s = 0; s < TT; s++) Pw[t * TT + s] *= inv;
        }
        __syncthreads();

        // out = P @ v  (N padded 8 -> 16)
        #pragma unroll
        for (int i = 0; i < 3; i++) {
            v8f acc;
            #pragma unroll
            for (int e = 0; e < 8; e++) acc[e] = 0.0f;
            for (int kk = 0; kk < TT; kk += 4) {
                const int k0 = kk + hi * 2;
                v2f a; a.x = Pw[(16 * i + row) * TT + k0]; a.y = Pw[(16 * i + row) * TT + k0 + 1];
                v2f bb;
                if (row < 8) { bb.x = vh[k0 * KH + row]; bb.y = vh[(k0 + 1) * KH + row]; }
                else         { bb.x = 0.0f; bb.y = 0.0f; }
                acc = wmma_f32(a, bb, acc);
            }
            if (row < 8) {
                #pragma unroll
                for (int r = 0; r < 8; r++) {
                    const int t = 16 * i + r + hi * 8;
                    const size_t m = ((size_t)(b * TT + t) * NN + n) * 64;
                    yatt[m + h * 8 + row] = acc[r];
                }
            }
        }
        __syncthreads();
    }
}

// ---------------- stage 4: z = att @ Wo^T, per-channel stats ----------------
__global__ __launch_bounds__(128)
void fc_out_kernel(const float* __restrict__ yatt, const float* __restrict__ Wo,
                   float* __restrict__ z, float* __restrict__ stat)
{
    __shared__ float red[128];
    const int tid  = threadIdx.x;
    const int wave = tid >> 5;
    const int lane = tid & 31;
    red[tid] = 0.0f;
    __syncthreads();

    const int m0    = (blockIdx.x * 4 + wave) * 16;
    const int row   = lane & 15;
    const int khalf = (lane >> 4) * 2;
    const int rowhi = (lane >> 4) * 8;

    v8f acc[4];
    #pragma unroll
    for (int i = 0; i < 4; i++)
        #pragma unroll
        for (int e = 0; e < 8; e++) acc[i][e] = 0.0f;

    const size_t mrow = (size_t)(m0 + row);
    for (int kk = 0; kk < 64; kk += 4) {
        const float* src = yatt + mrow * 64 + kk + khalf;
        v2f a; a.x = src[0]; a.y = src[1];
        #pragma unroll
        for (int nt = 0; nt < 4; nt++) {
            const int o = nt * 16 + row;
            const float* wp = Wo + o * 64 + kk + khalf;
            v2f b; b.x = wp[0]; b.y = wp[1];
            acc[nt] = wmma_f32(a, b, acc[nt]);
        }
    }

    #pragma unroll
    for (int nt = 0; nt < 4; nt++) {
        const int col = nt * 16 + row;
        float s = 0.0f, ss = 0.0f;
        #pragma unroll
        for (int r = 0; r < 8; r++) {
            const float v = acc[nt][r];
            z[(size_t)(m0 + r + rowhi) * 64 + col] = v;
            s += v; ss += v * v;
        }
        atomicAdd(&red[col], s);
        atomicAdd(&red[64 + col], ss);
    }
    __syncthreads();
    atomicAdd(&stat[768 + tid], red[tid]);
}

// ---------------- stage 5: out = relu(a*z + c) ----------------
__global__ void apply_out_kernel(const float* __restrict__ z, const float* __restrict__ stat,
                                 float* __restrict__ out)
{
    const size_t i = (size_t)blockIdx.x * 256 + threadIdx.x;
    if (i < (size_t)MTOT * 64) {
        const int ch = (int)(i & 63);
        out[i] = fmaxf(0.0f, stat[896 + ch] * z[i] + stat[960 + ch]);
    }
}

extern "C" void kernel_launch(void* const* d_in, const int* in_sizes, int n_in,
                              void* d_out, int out_size, void* d_ws, size_t ws_size,
                              hipStream_t stream) {
    const float* X     = (const float*)d_in[0];
    const float* STE   = (const float*)d_in[1];
    const float* Wq    = (const float*)d_in[2];
    const float* gq    = (const float*)d_in[4];
    const float* betaq = (const float*)d_in[5];
    const float* Wk    = (const float*)d_in[6];
    const float* gk    = (const float*)d_in[8];
    const float* betak = (const float*)d_in[9];
    const float* Wv    = (const float*)d_in[10];
    const float* gv    = (const float*)d_in[12];
    const float* betav = (const float*)d_in[13];
    const float* Wo    = (const float*)d_in[14];
    const float* go    = (const float*)d_in[16];
    const float* betao = (const float*)d_in[17];
    float* out = (float*)d_out;

    float* wsf = (float*)d_ws;
    const size_t S = (size_t)MTOT * 64;          // 24,576,000 floats per buffer
    float* yq   = wsf;
    float* yk   = wsf + S;
    float* yv   = wsf + 2 * S;
    float* stat = wsf + 3 * S;                   // 1024 floats
    float* yatt = yq;                            // safe alias (block-local read-then-write)
    float* z    = yk;                            // yk dead after attention

    const float invM = 1.0f / (float)MTOT;

    zero_stats_kernel<<<1, 256, 0, stream>>>(stat);
    fc_qkv_kernel<<<MTOT / 64, 128, 0, stream>>>(X, STE, Wq, Wk, Wv, yq, yk, yv, stat);
    bn_finalize_kernel<<<1, 64, 0, stream>>>(stat,       stat + 192, gq, betaq, stat + 384, stat + 576, invM);
    bn_finalize_kernel<<<1, 64, 0, stream>>>(stat + 64,  stat + 256, gk, betak, stat + 448, stat + 640, invM);
    bn_finalize_kernel<<<1, 64, 0, stream>>>(stat + 128, stat + 320, gv, betav, stat + 512, stat + 704, invM);
    attn_kernel<<<BB * NN, 64, 0, stream>>>(yq, yk, yv, stat, yatt);
    fc_out_kernel<<<MTOT / 64, 128, 0, stream>>>(yatt, Wo, z, stat);
    bn_finalize_kernel<<<1, 64, 0, stream>>>(stat + 768, stat + 832, go, betao, stat + 896, stat + 960, invM);
    apply_out_kernel<<<(MTOT * 64) / 256, 256, 0, stream>>>(z, stat, out);
}